// TitanMemory_37950331027672
// MI455X (gfx1250) — compile-verified
//
#include <hip/hip_runtime.h>
#include <hip/hip_bf16.h>
#include <math.h>

// ---------------- problem constants (match reference) ----------------
#define SEQ     16384
#define D       1024
#define CHUNK   128
#define NCHUNK  (SEQ / CHUNK)        // 128
#define FORGET_RATE   0.1f
#define SURPRISE_LR   0.01f
#define MOMENTUM      0.9f
#define INV_C         (1.0f / (float)CHUNK)

typedef float v2f __attribute__((ext_vector_type(2)));
typedef float v8f __attribute__((ext_vector_type(8)));

#define WMMA_F32(A, B, C) \
    __builtin_amdgcn_wmma_f32_16x16x4_f32(false, (A), false, (B), (short)0, (C), false, false)

// fast silu: x * sigmoid(x), sigmoid via v_rcp_f32 (~1 ulp)
__device__ __forceinline__ float silu_f(float x) {
    return x * __builtin_amdgcn_rcpf(1.0f + __expf(-x));
}

// =====================================================================
// Phase 1a: Y = silu(X @ W^T)  X:[rows,D] row-major, W:[D,D] row-major.
// 32x32 output tile per wave (2 A-frags x 2 B-frags, 4 accumulators)
// -> 8 FLOP/byte of fresh operands. block = 256 (8 waves), each wave a
// distinct 32-wide N stripe; grid (D/256, rows/32).
// =====================================================================
__global__ __launch_bounds__(256)
void proj_silu_kernel(const float* __restrict__ X,
                      const float* __restrict__ W,
                      float* __restrict__ Y) {
    const int lane = threadIdx.x & 31;
    const int wave = threadIdx.x >> 5;
    const int n0   = (blockIdx.x * 8 + wave) * 32;
    const int m0   = blockIdx.y * 32;
    const int half = lane >> 4;          // 0 or 1
    const int lid  = lane & 15;
    const int koff = half * 2;

    const float* a0p = X + (size_t)(m0 + lid)      * D;
    const float* a1p = X + (size_t)(m0 + 16 + lid) * D;
    const float* b0p = W + (size_t)(n0 + lid)      * D;
    const float* b1p = W + (size_t)(n0 + 16 + lid) * D;

    v8f acc00 = {}, acc01 = {}, acc10 = {}, acc11 = {};
    #pragma unroll 2
    for (int kk = 0; kk < D; kk += 4) {
        v2f a0 = *(const v2f*)(a0p + kk + koff);
        v2f a1 = *(const v2f*)(a1p + kk + koff);
        v2f b0 = *(const v2f*)(b0p + kk + koff);
        v2f b1 = *(const v2f*)(b1p + kk + koff);
        acc00 = WMMA_F32(a0, b0, acc00);
        acc01 = WMMA_F32(a0, b1, acc01);
        acc10 = WMMA_F32(a1, b0, acc10);
        acc11 = WMMA_F32(a1, b1, acc11);
    }

    const int rbase = half * 8;
    #pragma unroll
    for (int r = 0; r < 8; ++r) {
        const size_t row0 = (size_t)(m0 + rbase + r) * D;
        const size_t row1 = row0 + (size_t)16 * D;
        Y[row0 + n0 + lid]      = silu_f(acc00[r]);
        Y[row0 + n0 + 16 + lid] = silu_f(acc01[r]);
        Y[row1 + n0 + lid]      = silu_f(acc10[r]);
        Y[row1 + n0 + 16 + lid] = silu_f(acc11[r]);
    }
}

// =====================================================================
// Phase 1b: row-wise l2 normalization (+ optional dot with Wf -> kwf)
// one wave per row; wave32 butterfly reduction
// =====================================================================
__global__ __launch_bounds__(256)
void rownorm_kernel(float* __restrict__ Y,
                    const float* __restrict__ wf,   // may be null
                    float* __restrict__ kwf,        // may be null
                    int rows) {
    const int lane = threadIdx.x & 31;
    const int wave = threadIdx.x >> 5;
    const int row  = blockIdx.x * 8 + wave;
    if (row >= rows) return;
    float* yrow = Y + (size_t)row * D;

    float ss = 0.0f;
    #pragma unroll 4
    for (int h = lane; h < D; h += 32) {
        float y = yrow[h];
        ss += y * y;
    }
    #pragma unroll
    for (int m = 16; m >= 1; m >>= 1) ss += __shfl_xor(ss, m, 32);

    const float scale = 1.0f / (sqrtf(ss) + 1e-8f);

    float d = 0.0f;
    #pragma unroll 4
    for (int h = lane; h < D; h += 32) {
        float y = yrow[h] * scale;
        yrow[h] = y;
        if (wf) d += y * wf[h];
    }
    if (wf) {
        #pragma unroll
        for (int m = 16; m >= 1; m >>= 1) d += __shfl_xor(d, m, 32);
        if (lane == 0) kwf[row] = d;
    }
}

// =====================================================================
// Phase 1c: alphas[c] = sigmoid(mean(kwf[c*CHUNK : (c+1)*CHUNK])) * rate
// =====================================================================
__global__ __launch_bounds__(128)
void alpha_kernel(const float* __restrict__ kwf, float* __restrict__ alphas) {
    const int c = threadIdx.x;   // one block of NCHUNK threads
    float s = 0.0f;
    const float* p = kwf + c * CHUNK;
    #pragma unroll 8
    for (int i = 0; i < CHUNK; ++i) s += p[i];
    const float mean = s * INV_C;
    alphas[c] = FORGET_RATE * __builtin_amdgcn_rcpf(1.0f + __expf(-mean));
}

// =====================================================================
// Recurrent step 1: err = Kc @ M - Vc    [CHUNK x D], K-dim = D
// 16x16/wave (keeps 512 waves of parallelism on the serial chain);
// prefetch the strided M-row stream ahead (global_prefetch_b8).
// grid (D/128, CHUNK/16), block 256
// =====================================================================
__global__ __launch_bounds__(256)
void err_kernel(const float* __restrict__ Kc,
                const float* __restrict__ Vc,
                const float* __restrict__ M,
                float* __restrict__ err) {
    const int lane = threadIdx.x & 31;
    const int wave = threadIdx.x >> 5;
    const int n0   = (blockIdx.x * 8 + wave) * 16;
    const int m0   = blockIdx.y * 16;
    const int half = lane >> 4;
    const int lid  = lane & 15;
    const int koff = half * 2;
    const int ncol = n0 + lid;

    const float* arow = Kc + (size_t)(m0 + lid) * D;

    v8f acc = {};
    #pragma unroll 4
    for (int kk = 0; kk < D; kk += 4) {
        v2f a = *(const v2f*)(arow + kk + koff);
        v2f b;
        b.x = M[(size_t)(kk + koff)     * D + ncol];
        b.y = M[(size_t)(kk + koff + 1) * D + ncol];
        if (kk + 16 < D)
            __builtin_prefetch(M + (size_t)(kk + 16 + koff) * D + ncol, 0, 1);
        acc = WMMA_F32(a, b, acc);
    }

    const int rbase = half * 8;
    #pragma unroll
    for (int r = 0; r < 8; ++r) {
        const int row = m0 + rbase + r;
        err[(size_t)row * D + ncol] =
            acc[r] - Vc[(size_t)row * D + ncol];
    }
}

// =====================================================================
// Recurrent step 2: G = Kc^T @ err  (K-dim = CHUNK)
//   S = MOMENTUM*S - SURPRISE_LR * (G/CHUNK)
//   M = (1 - alpha_c)*M + S
// 32x32 per wave (1024x1024 output -> 1024 waves); grid (D/256, D/32)
// =====================================================================
__global__ __launch_bounds__(256)
void update_kernel(const float* __restrict__ Kc,
                   const float* __restrict__ err,
                   float* __restrict__ M,
                   float* __restrict__ S,
                   const float* __restrict__ alphas,
                   int chunk) {
    const int lane = threadIdx.x & 31;
    const int wave = threadIdx.x >> 5;
    const int n0   = (blockIdx.x * 8 + wave) * 32;
    const int m0   = blockIdx.y * 32;
    const int half = lane >> 4;
    const int lid  = lane & 15;
    const int koff = half * 2;
    const int nc0  = n0 + lid;
    const int nc1  = n0 + 16 + lid;
    const int mg0  = m0 + lid;            // A = Kc^T : A[m][k] = Kc[k][m]
    const int mg1  = m0 + 16 + lid;

    v8f acc00 = {}, acc01 = {}, acc10 = {}, acc11 = {};
    #pragma unroll 4
    for (int kk = 0; kk < CHUNK; kk += 4) {
        const size_t r0 = (size_t)(kk + koff)     * D;
        const size_t r1 = (size_t)(kk + koff + 1) * D;
        v2f a0, a1, b0, b1;
        a0.x = Kc[r0 + mg0];  a0.y = Kc[r1 + mg0];
        a1.x = Kc[r0 + mg1];  a1.y = Kc[r1 + mg1];
        b0.x = err[r0 + nc0]; b0.y = err[r1 + nc0];
        b1.x = err[r0 + nc1]; b1.y = err[r1 + nc1];
        acc00 = WMMA_F32(a0, b0, acc00);
        acc01 = WMMA_F32(a0, b1, acc01);
        acc10 = WMMA_F32(a1, b0, acc10);
        acc11 = WMMA_F32(a1, b1, acc11);
    }

    const float alpha  = alphas[chunk];
    const float one_ma = 1.0f - alpha;
    const int   rbase  = half * 8;
    #pragma unroll
    for (int r = 0; r < 8; ++r) {
        const size_t row0 = (size_t)(m0 + rbase + r) * D;
        const size_t row1 = row0 + (size_t)16 * D;
        const size_t idx[4] = { row0 + nc0, row0 + nc1, row1 + nc0, row1 + nc1 };
        const float  g[4]   = { acc00[r], acc01[r], acc10[r], acc11[r] };
        #pragma unroll
        for (int t = 0; t < 4; ++t) {
            const float s = MOMENTUM * S[idx[t]] - SURPRISE_LR * (g[t] * INV_C);
            S[idx[t]] = s;
            M[idx[t]] = one_ma * M[idx[t]] + s;
        }
    }
}

// =====================================================================
// Recurrent step 3: out_c = Qc @ M   [CHUNK x D], K-dim = D
// =====================================================================
__global__ __launch_bounds__(256)
void retrieve_kernel(const float* __restrict__ Qc,
                     const float* __restrict__ M,
                     float* __restrict__ out) {
    const int lane = threadIdx.x & 31;
    const int wave = threadIdx.x >> 5;
    const int n0   = (blockIdx.x * 8 + wave) * 16;
    const int m0   = blockIdx.y * 16;
    const int half = lane >> 4;
    const int lid  = lane & 15;
    const int koff = half * 2;
    const int ncol = n0 + lid;

    const float* arow = Qc + (size_t)(m0 + lid) * D;

    v8f acc = {};
    #pragma unroll 4
    for (int kk = 0; kk < D; kk += 4) {
        v2f a = *(const v2f*)(arow + kk + koff);
        v2f b;
        b.x = M[(size_t)(kk + koff)     * D + ncol];
        b.y = M[(size_t)(kk + koff + 1) * D + ncol];
        if (kk + 16 < D)
            __builtin_prefetch(M + (size_t)(kk + 16 + koff) * D + ncol, 0, 1);
        acc = WMMA_F32(a, b, acc);
    }

    const int rbase = half * 8;
    #pragma unroll
    for (int r = 0; r < 8; ++r) {
        const int row = m0 + rbase + r;
        out[(size_t)row * D + ncol] = acc[r];
    }
}

// =====================================================================
// host-side orchestration
// =====================================================================
extern "C" void kernel_launch(void* const* d_in, const int* in_sizes, int n_in,
                              void* d_out, int out_size, void* d_ws, size_t ws_size,
                              hipStream_t stream) {
    const float* x  = (const float*)d_in[0];   // [SEQ, D]
    const float* Wk = (const float*)d_in[1];   // [D, D]
    const float* Wv = (const float*)d_in[2];   // [D, D]
    const float* Wq = (const float*)d_in[3];   // [D, D]
    const float* Wf = (const float*)d_in[4];   // [1, D]
    float* out = (float*)d_out;                // [SEQ, D]

    // workspace layout (floats)
    float* ws     = (float*)d_ws;
    const size_t SD = (size_t)SEQ * D;         // 16M floats
    const size_t DD = (size_t)D * D;           // 1M floats
    float* Kmat   = ws;                        // [SEQ, D]
    float* Qmat   = Kmat + SD;                 // [SEQ, D]
    float* Vmat   = Qmat + SD;                 // [SEQ, D]
    float* Mst    = Vmat + SD;                 // [D, D]
    float* Sst    = Mst + DD;                  // [D, D]
    float* errb   = Sst + DD;                  // [CHUNK, D]
    float* kwf    = errb + (size_t)CHUNK * D;  // [SEQ]
    float* alphas = kwf + SEQ;                 // [NCHUNK]

    // zero the recurrent state every call (graph-capture-safe memset nodes)
    hipMemsetAsync(Mst, 0, DD * sizeof(float), stream);
    hipMemsetAsync(Sst, 0, DD * sizeof(float), stream);

    // ---- Phase 1: parallel projections over the whole sequence ----
    {
        dim3 grid(D / 256, SEQ / 32);          // (4, 512)
        dim3 block(256);
        proj_silu_kernel<<<grid, block, 0, stream>>>(x, Wk, Kmat);
        proj_silu_kernel<<<grid, block, 0, stream>>>(x, Wq, Qmat);
        proj_silu_kernel<<<grid, block, 0, stream>>>(x, Wv, Vmat);
    }
    {
        dim3 grid((SEQ + 7) / 8);              // 8 rows per block (1 wave/row)
        dim3 block(256);
        rownorm_kernel<<<grid, block, 0, stream>>>(Kmat, Wf, kwf, SEQ);
        rownorm_kernel<<<grid, block, 0, stream>>>(Qmat, nullptr, nullptr, SEQ);
    }
    alpha_kernel<<<1, NCHUNK, 0, stream>>>(kwf, alphas);

    // ---- Phase 2: serial recurrence over chunks ----
    dim3 gridE(D / 128, CHUNK / 16);           // (8, 8)
    dim3 gridU(D / 256, D / 32);               // (4, 32)
    dim3 block(256);
    for (int c = 0; c < NCHUNK; ++c) {
        const float* Kc = Kmat + (size_t)c * CHUNK * D;
        const float* Vc = Vmat + (size_t)c * CHUNK * D;
        const float* Qc = Qmat + (size_t)c * CHUNK * D;
        float*       oc = out  + (size_t)c * CHUNK * D;

        err_kernel<<<gridE, block, 0, stream>>>(Kc, Vc, Mst, errb);
        update_kernel<<<gridU, block, 0, stream>>>(Kc, errb, Mst, Sst, alphas, c);
        retrieve_kernel<<<gridE, block, 0, stream>>>(Qc, Mst, oc);
    }
}